// TDGCN_52682068852781
// MI455X (gfx1250) — compile-verified
//
#include <hip/hip_runtime.h>

// ============================================================================
// Two-layer GCN for MI455X (gfx1250, wave32).
//   out = relu( S (relu( S (X W1) + b1 )) W2 + b2 ),  S = D^-1/2 (A+I) D^-1/2
// Dense GEMMs use V_WMMA_F32_16X16X4_F32 (one wave per 16x16 tile).
// Aggregation = L2-resident float atomics (whole working set < 192MB L2).
// ============================================================================

typedef __attribute__((ext_vector_type(2))) float v2f;
typedef __attribute__((ext_vector_type(8))) float v8f;

// ---- degree = 1 (self loop) + count of col ---------------------------------
__global__ void init_deg_kernel(float* __restrict__ deg, int n) {
  int i = blockIdx.x * blockDim.x + threadIdx.x;
  if (i < n) deg[i] = 1.0f;  // self-loop contribution
}

__global__ void count_deg_kernel(const long long* __restrict__ col,
                                 float* __restrict__ deg, long long E) {
  long long e = (long long)blockIdx.x * blockDim.x + threadIdx.x;
  if (e < E) atomicAdd(&deg[col[e]], 1.0f);
}

__global__ void dinv_kernel(float* __restrict__ d, int n) {
  int i = blockIdx.x * blockDim.x + threadIdx.x;
  if (i < n) {
    float v = d[i];
    d[i] = (v > 0.0f) ? rsqrtf(v) : 0.0f;
  }
}

// ---- WMMA GEMM: [nrows x K] @ [K x NOUT], epilogue scales by dinv[row] and
//      writes the result to BOTH y (edge-gather source) and agg (accumulator
//      pre-seeded with the self-loop term). One wave per 16x16 tile. ---------
template <int K, int NOUT>
__global__ void gemm_scale_dup_kernel(const float* __restrict__ X,
                                      const float* __restrict__ W,
                                      const float* __restrict__ dinv,
                                      float* __restrict__ y,
                                      float* __restrict__ agg, int nrows) {
  const int lane  = threadIdx.x & 31;
  const int wave  = (blockIdx.x * blockDim.x + threadIdx.x) >> 5;
  const int tilesN = NOUT / 16;
  const int tm = wave / tilesN;
  const int tn = wave % tilesN;
  const int tilesM = (nrows + 15) >> 4;
  if (tm >= tilesM) return;  // wave-uniform: EXEC stays all-1s for WMMA

  const int row0 = tm << 4;
  const int col0 = tn << 4;
  const int m    = lane & 15;   // A row / B-and-C column within tile
  const int half = lane >> 4;   // K sub-pair select (A/B), M+8 select (C/D)

  const float* __restrict__ xrow = X + (size_t)(row0 + m) * K;
  const float* __restrict__ wcol = W + col0 + m;

  v8f c = {};
#pragma unroll
  for (int k0 = 0; k0 < K; k0 += 4) {
    const int ka = k0 + half * 2;
    // A 16x4 f32: lanes 0-15 -> K={k0,k0+1}, lanes 16-31 -> K={k0+2,k0+3}
    v2f a;
    a.x = xrow[ka];
    a.y = xrow[ka + 1];
    // B 4x16 f32: row-striped across lanes per VGPR, halves split K pairs
    v2f b;
    b.x = wcol[(size_t)(ka)     * NOUT];
    b.y = wcol[(size_t)(ka + 1) * NOUT];
    c = __builtin_amdgcn_wmma_f32_16x16x4_f32(
        /*neg_a=*/false, a, /*neg_b=*/false, b,
        /*c_mod=*/(short)0, c, /*reuse_a=*/false, /*reuse_b=*/false);
  }

  // C/D layout: VGPR r, lanes 0-15 -> M=r, lanes 16-31 -> M=r+8; N = lane&15
#pragma unroll
  for (int r = 0; r < 8; ++r) {
    const int mm = row0 + r + (half << 3);
    const float v = c[r] * dinv[mm];
    const size_t idx = (size_t)mm * NOUT + col0 + m;
    y[idx]   = v;   // gather source for the edge scatter
    agg[idx] = v;   // accumulator seeded with self-loop term
  }
}

// ---- edge scatter: agg[col] += y[row], 4 features per thread (float4) ------
template <int F>
__global__ void scatter_kernel(const long long* __restrict__ row,
                               const long long* __restrict__ col,
                               const float* __restrict__ y,
                               float* __restrict__ agg, long long E) {
  constexpr int G  = F / 4;              // float4 groups per edge
  constexpr int LG = (G == 16) ? 4 : 3;  // log2(G)
  long long tid = (long long)blockIdx.x * blockDim.x + threadIdx.x;
  if (tid >= E * (long long)G) return;
  const int g       = (int)(tid & (G - 1));
  const long long e = tid >> LG;
  const long long r = row[e];
  const long long cd = col[e];
  const float4 v = *(const float4*)(y + r * F + g * 4);
  float* dst = agg + cd * F + g * 4;
  atomicAdd(dst + 0, v.x);
  atomicAdd(dst + 1, v.y);
  atomicAdd(dst + 2, v.z);
  atomicAdd(dst + 3, v.w);
}

// ---- finalize: out = relu(agg * dinv[node] + b[f]) (may run in-place) ------
template <int F>
__global__ void finalize_kernel(const float* __restrict__ agg,
                                const float* __restrict__ dinv,
                                const float* __restrict__ b,
                                float* __restrict__ out, int n) {
  constexpr int LF = (F == 64) ? 6 : 5;
  long long tid = (long long)blockIdx.x * blockDim.x + threadIdx.x;
  if (tid >= (long long)n * F) return;
  const int node = (int)(tid >> LF);
  const int f    = (int)(tid & (F - 1));
  const float v  = agg[tid] * dinv[node] + b[f];
  out[tid] = (v > 0.0f) ? v : 0.0f;
}

// ============================================================================
extern "C" void kernel_launch(void* const* d_in, const int* in_sizes, int n_in,
                              void* d_out, int out_size, void* d_ws,
                              size_t ws_size, hipStream_t stream) {
  const float* x  = (const float*)d_in[0];   // [N,128]
  const float* W1 = (const float*)d_in[1];   // [128,64]
  const float* b1 = (const float*)d_in[2];   // [64]
  const float* W2 = (const float*)d_in[3];   // [64,32]
  const float* b2 = (const float*)d_in[4];   // [32]
  const long long* eidx = (const long long*)d_in[5];  // [2,E] int64

  const int N       = in_sizes[0] / 128;
  const long long E = (long long)in_sizes[5] / 2;
  const long long* row = eidx;       // source nodes
  const long long* col = eidx + E;   // destination nodes

  float* ws   = (float*)d_ws;
  float* dinv = ws;                       // N floats (deg -> rsqrt in place)
  float* y1   = ws + N;                   // N*64 (reused as h after layer 1)
  float* agg1 = y1 + (size_t)N * 64;      // N*64 (reused as y2 in layer 2)
  float* outp = (float*)d_out;            // N*32, doubles as agg2

  const int B = 256;

  // degrees -> dinv
  init_deg_kernel<<<(N + B - 1) / B, B, 0, stream>>>(dinv, N);
  count_deg_kernel<<<(unsigned)((E + B - 1) / B), B, 0, stream>>>(col, dinv, E);
  dinv_kernel<<<(N + B - 1) / B, B, 0, stream>>>(dinv, N);

  // ---- layer 1: [N,128]@[128,64] -> scatter -> relu ----
  {
    const int tilesM = (N + 15) / 16;
    const long long threads = (long long)tilesM * (64 / 16) * 32;
    gemm_scale_dup_kernel<128, 64>
        <<<(unsigned)((threads + B - 1) / B), B, 0, stream>>>(x, W1, dinv, y1,
                                                              agg1, N);
  }
  scatter_kernel<64><<<(unsigned)((E * 16 + B - 1) / B), B, 0, stream>>>(
      row, col, y1, agg1, E);
  finalize_kernel<64>
      <<<(unsigned)(((long long)N * 64 + B - 1) / B), B, 0, stream>>>(
          agg1, dinv, b1, /*h=*/y1, N);

  // ---- layer 2: [N,64]@[64,32] -> scatter -> relu (agg2 lives in d_out) ----
  {
    const int tilesM = (N + 15) / 16;
    const long long threads = (long long)tilesM * (32 / 16) * 32;
    gemm_scale_dup_kernel<64, 32>
        <<<(unsigned)((threads + B - 1) / B), B, 0, stream>>>(
            /*X=h*/ y1, W2, dinv, /*y2=*/agg1, /*agg2=*/outp, N);
  }
  scatter_kernel<32><<<(unsigned)((E * 8 + B - 1) / B), B, 0, stream>>>(
      row, col, agg1, outp, E);
  finalize_kernel<32>
      <<<(unsigned)(((long long)N * 32 + B - 1) / B), B, 0, stream>>>(
          outp, dinv, b2, outp, N);
}